// GatedGCNLayer_25898652795468
// MI455X (gfx1250) — compile-verified
//
#include <hip/hip_runtime.h>
#include <math.h>
#include <stdint.h>

typedef __attribute__((ext_vector_type(2))) float v2f;
typedef __attribute__((ext_vector_type(8))) float v8f;

#define DFEAT 64
#define LDSSTRIDE 68   // floats; row padding -> conflict-free ds_load_b64 A-frag reads
#define BN_EPS 1e-5f

// ---------------------------------------------------------------------------
// Kernel 1: fused 4-way GEMM  out = x @ W^T + b  for W in {Wk,Wq,Wv,Ws}
// block = 256 threads (8 wave32), handles 32 node rows.
// wave w: matrix = w & 3 (0->k, 1->q, 2->v, 3->s written into h), mtile = w>>2.
// Each wave: 4 N-tiles of 16 cols, K=64 in 16 steps of v_wmma_f32_16x16x4_f32.
// x tile staged to LDS via CDNA5 async global->LDS copies (ASYNCcnt path).
// ---------------------------------------------------------------------------
__global__ __launch_bounds__(256)
void gemm4_wmma(const float* __restrict__ x,
                const float* __restrict__ Wk, const float* __restrict__ bk,
                const float* __restrict__ Wq, const float* __restrict__ bq,
                const float* __restrict__ Wv, const float* __restrict__ bv,
                const float* __restrict__ Ws, const float* __restrict__ bs,
                float* __restrict__ ko, float* __restrict__ qo,
                float* __restrict__ vo, float* __restrict__ ho,
                int nrows)
{
    __shared__ float xs[32 * LDSSTRIDE];

    const int tid   = threadIdx.x;
    const int mbase = blockIdx.x * 32;

    // Stage x[mbase .. mbase+31, 0..63] into LDS. 256 thr * 8 floats = 32*64.
    // Uses gfx1250 async copy: global_load_async_to_lds_b128 (no VGPR bounce,
    // tracked with ASYNCcnt; wait + workgroup barrier before ds_load reads).
    {
        const int r  = tid >> 3;          // 0..31 (8 threads per row)
        const int c0 = (tid & 7) * 8;     // 0,8,...,56
        int rr = mbase + r;
        if (rr >= nrows) rr = nrows - 1;  // clamp; keeps EXEC all-1s later
        const float* src = x + (size_t)rr * DFEAT + c0;
        unsigned lds0 = (unsigned)(uintptr_t)(&xs[r * LDSSTRIDE + c0]);
        asm volatile(
            "global_load_async_to_lds_b128 %0, %2, off\n\t"
            "global_load_async_to_lds_b128 %1, %3, off"
            :
            : "v"(lds0), "v"(lds0 + 16u), "v"(src), "v"(src + 4)
            : "memory");
        asm volatile("s_wait_asynccnt 0x0" ::: "memory");
    }
    __syncthreads();

    const int wave  = tid >> 5;
    const int lane  = tid & 31;
    const int mat   = wave & 3;
    const int mtile = wave >> 2;          // 0 or 1

    const float* W = (mat == 0) ? Wk : (mat == 1) ? Wq : (mat == 2) ? Wv : Ws;
    const float* b = (mat == 0) ? bk : (mat == 1) ? bq : (mat == 2) ? bv : bs;
    float*     out = (mat == 0) ? ko : (mat == 1) ? qo : (mat == 2) ? vo : ho;

    const int nlane = lane & 15;
    const int koff  = (lane >> 4) * 2;    // lanes 0-15: K+0/K+1, lanes 16-31: K+2/K+3
    const int arow  = mtile * 16 + nlane; // A-matrix row for this lane

    v8f acc[4];
    acc[0] = (v8f){}; acc[1] = (v8f){}; acc[2] = (v8f){}; acc[3] = (v8f){};

#pragma unroll
    for (int s = 0; s < 16; ++s) {
        const int kk = 4 * s + koff;
        v2f a;
        a.x = xs[arow * LDSSTRIDE + kk];
        a.y = xs[arow * LDSSTRIDE + kk + 1];
#pragma unroll
        for (int nt = 0; nt < 4; ++nt) {
            const int n = nt * 16 + nlane;        // output column (W row)
            const float* wp = W + (size_t)n * DFEAT + kk;
            v2f bb;
            bb.x = wp[0];
            bb.y = wp[1];
            acc[nt] = __builtin_amdgcn_wmma_f32_16x16x4_f32(
                false, a, false, bb, (short)0, acc[nt], false, false);
        }
    }

    // Store D tiles: VGPR r of acc holds rows {r, r+8} (lane<16 / lane>=16).
    const int row0 = mbase + mtile * 16 + ((lane >> 4) * 8);
#pragma unroll
    for (int nt = 0; nt < 4; ++nt) {
        const int col   = nt * 16 + nlane;
        const float bias = b[col];
#pragma unroll
        for (int r = 0; r < 8; ++r) {
            const int row = row0 + r;
            if (row < nrows)
                out[(size_t)row * DFEAT + col] = acc[nt][r] + bias;
        }
    }
}

// ---------------------------------------------------------------------------
// Kernel 2: per-edge gated message + scatter-add.
// 16 lanes per edge, float4 per lane (16*4 = 64 features).
// h must already contain lin_skip(x) (written by gemm4_wmma).
// ---------------------------------------------------------------------------
__global__ __launch_bounds__(256)
void edge_gate(const int* __restrict__ ei, int nedges,
               const float* __restrict__ k, const float* __restrict__ q,
               const float* __restrict__ v, float* __restrict__ h)
{
    const long long g = (long long)blockIdx.x * blockDim.x + threadIdx.x;
    const long long e = g >> 4;
    if (e >= nedges) return;
    const int lane16 = (int)(g & 15);

    const int src = ei[e];                        // source j
    const int dst = ei[(long long)nedges + e];    // target i

    const size_t so = (size_t)src * DFEAT + lane16 * 4;
    const size_t to = (size_t)dst * DFEAT + lane16 * 4;

    float4 kd = *(const float4*)(k + to);
    float4 qs = *(const float4*)(q + so);
    float4 vs = *(const float4*)(v + so);

    float4 m;
    m.x = vs.x / (1.0f + __expf(-(kd.x + qs.x)));
    m.y = vs.y / (1.0f + __expf(-(kd.y + qs.y)));
    m.z = vs.z / (1.0f + __expf(-(kd.z + qs.z)));
    m.w = vs.w / (1.0f + __expf(-(kd.w + qs.w)));

    float* hp = h + to;
    atomicAdd(hp + 0, m.x);
    atomicAdd(hp + 1, m.y);
    atomicAdd(hp + 2, m.z);
    atomicAdd(hp + 3, m.w);
}

// ---------------------------------------------------------------------------
// Kernel 3: per-feature sum / sumsq for BatchNorm (train-mode batch stats).
// stats[0:64] = sum, stats[64:128] = sumsq. Must be zeroed before launch.
// ---------------------------------------------------------------------------
__global__ __launch_bounds__(256)
void bn_stats(const float* __restrict__ h, float* __restrict__ stats, int nrows)
{
    const int g     = blockIdx.x * blockDim.x + threadIdx.x;
    const int col   = g & 63;
    const int rbeg  = g >> 6;
    const int rstep = (gridDim.x * blockDim.x) >> 6;

    float s = 0.0f, s2 = 0.0f;
    for (int r = rbeg; r < nrows; r += rstep) {
        const float* p = h + (size_t)r * DFEAT + col;
        __builtin_prefetch(h + (size_t)(r + rstep) * DFEAT + col, 0, 0);
        float val = *p;
        s  += val;
        s2 += val * val;
    }
    atomicAdd(&stats[col], s);
    atomicAdd(&stats[64 + col], s2);
}

// ---------------------------------------------------------------------------
// Kernel 4: BN normalize + affine + ReLU + residual; float4 per thread.
// ---------------------------------------------------------------------------
__global__ __launch_bounds__(256)
void bn_final(const float* __restrict__ h, const float* __restrict__ x,
              const float* __restrict__ stats,
              const float* __restrict__ gamma, const float* __restrict__ beta,
              float* __restrict__ out, int nrows)
{
    const long long g = (long long)blockIdx.x * blockDim.x + threadIdx.x;
    const long long total = (long long)nrows * DFEAT;
    const long long e = g * 4;
    if (e >= total) return;

    const int col = (int)(e & 63);    // 4 consecutive feature columns
    const float invN = 1.0f / (float)nrows;

    float4 hv = *(const float4*)(h + e);
    float4 xv = *(const float4*)(x + e);
    float ov[4];
    float hvv[4] = {hv.x, hv.y, hv.z, hv.w};
    float xvv[4] = {xv.x, xv.y, xv.z, xv.w};

#pragma unroll
    for (int c = 0; c < 4; ++c) {
        const int cc   = col + c;
        const float mean = stats[cc] * invN;
        const float var  = stats[64 + cc] * invN - mean * mean;
        const float rs   = rsqrtf(var + BN_EPS);
        float val = (hvv[c] - mean) * rs * gamma[cc] + beta[cc];
        val = fmaxf(val, 0.0f);
        ov[c] = val + xvv[c];
    }
    float4 o = {ov[0], ov[1], ov[2], ov[3]};
    *(float4*)(out + e) = o;
}

// ---------------------------------------------------------------------------
// Host launch.
// Workspace layout (floats): k[N*64] | q[N*64] | v[N*64] | h[N*64] | stats[128]
// ---------------------------------------------------------------------------
extern "C" void kernel_launch(void* const* d_in, const int* in_sizes, int n_in,
                              void* d_out, int out_size, void* d_ws, size_t ws_size,
                              hipStream_t stream)
{
    const float* x     = (const float*)d_in[0];
    const int*   ei    = (const int*)d_in[1];
    const float* Wk    = (const float*)d_in[2];
    const float* bk    = (const float*)d_in[3];
    const float* Wq    = (const float*)d_in[4];
    const float* bq    = (const float*)d_in[5];
    const float* Wv    = (const float*)d_in[6];
    const float* bv    = (const float*)d_in[7];
    const float* Ws    = (const float*)d_in[8];
    const float* bs    = (const float*)d_in[9];
    const float* gamma = (const float*)d_in[10];
    const float* beta  = (const float*)d_in[11];

    const int nrows  = in_sizes[0] / DFEAT;   // N
    const int nedges = in_sizes[1] / 2;       // E

    char* ws = (char*)d_ws;
    const size_t nd = (size_t)nrows * DFEAT * sizeof(float);
    float* k     = (float*)(ws);
    float* q     = (float*)(ws + nd);
    float* v     = (float*)(ws + 2 * nd);
    float* h     = (float*)(ws + 3 * nd);
    float* stats = (float*)(ws + 4 * nd);

    hipMemsetAsync(stats, 0, 128 * sizeof(float), stream);

    // 1) k,q,v and h = lin_skip(x) via WMMA
    const int mblocks = (nrows + 31) / 32;
    gemm4_wmma<<<mblocks, 256, 0, stream>>>(x, Wk, bk, Wq, bq, Wv, bv, Ws, bs,
                                            k, q, v, h, nrows);

    // 2) gated edge messages, scatter-add into h
    const long long ethreads = (long long)nedges * 16;
    const int eblocks = (int)((ethreads + 255) / 256);
    edge_gate<<<eblocks, 256, 0, stream>>>(ei, nedges, k, q, v, h);

    // 3) batch-norm statistics
    bn_stats<<<512, 256, 0, stream>>>(h, stats, nrows);

    // 4) normalize + relu + residual -> d_out
    const long long total = (long long)nrows * DFEAT;
    const int fblocks = (int)((total / 4 + 255) / 256);
    bn_final<<<fblocks, 256, 0, stream>>>(h, x, stats, gamma, beta,
                                          (float*)d_out, nrows);
}